// GeMVoxel_64613488001756
// MI455X (gfx1250) — compile-verified
//
#include <hip/hip_runtime.h>
#include <hip/hip_bf16.h>

#define CDIM 256
#define OUT_DIM 256
#define NSEG 32
#define GEM_EPS 1e-6f

typedef float v2f __attribute__((ext_vector_type(2)));
typedef float v8f __attribute__((ext_vector_type(8)));

// ---------------------------------------------------------------------------
// Kernel 1: segment boundaries via binary search over sorted batch_indices.
// start[s] = first i with idx[i] >= s, for s = 0..32 (start[32] == N).
// ---------------------------------------------------------------------------
__global__ void k_boundaries(const int* __restrict__ idx, int N, int* __restrict__ start) {
    int t = threadIdx.x;
    if (t > NSEG) return;
    int lo = 0, hi = N;
    while (lo < hi) {
        int mid = (lo + hi) >> 1;
        if (idx[mid] < t) lo = mid + 1; else hi = mid;
    }
    start[t] = lo;
}

__device__ __forceinline__ float gem_pow(float v, float pv) {
    // max(v, EPS)^pv  via exp2(pv * log2(v));  v >= EPS > 0
    float m = fmaxf(v, GEM_EPS);
    return __builtin_amdgcn_exp2f(pv * __builtin_amdgcn_logf(m));
}

// ---------------------------------------------------------------------------
// Kernel 2: deterministic partial segment sums of x^p.
// grid = (P, NSEG), block = 256 (8 waves). Block (pb, s) owns rows j of
// segment s with (j mod 8P) in [8*pb, 8*pb+8); wave w takes j = 8*pb + w + t*8P
// in ascending order. Lane handles 8 columns (two float4 loads).
// Fixed-order LDS reduce of the 8 wave partials -> part[s][pb][256].
// ---------------------------------------------------------------------------
__global__ void k_partials(const float* __restrict__ x, const float* __restrict__ p,
                           const int* __restrict__ start, float* __restrict__ part, int P) {
    const int pb = blockIdx.x;
    const int s  = blockIdx.y;
    const int tid  = threadIdx.x;
    const int w    = tid >> 5;
    const int lane = tid & 31;

    const float pv = p[0];
    const int rs = start[s];
    const int L  = start[s + 1] - rs;
    const int stride = P << 3;

    float acc[8];
#pragma unroll
    for (int k = 0; k < 8; ++k) acc[k] = 0.0f;

    for (int j = (pb << 3) + w; j < L; j += stride) {
        const float4* rp = (const float4*)(x + (size_t)(rs + j) * CDIM);
        float4 a = rp[lane];        // cols 4*lane .. 4*lane+3
        float4 c = rp[lane + 32];   // cols 128+4*lane ..
        acc[0] += gem_pow(a.x, pv); acc[1] += gem_pow(a.y, pv);
        acc[2] += gem_pow(a.z, pv); acc[3] += gem_pow(a.w, pv);
        acc[4] += gem_pow(c.x, pv); acc[5] += gem_pow(c.y, pv);
        acc[6] += gem_pow(c.z, pv); acc[7] += gem_pow(c.w, pv);
    }

    __shared__ float red[8][CDIM];
#pragma unroll
    for (int k = 0; k < 4; ++k) {
        red[w][4 * lane + k]       = acc[k];
        red[w][128 + 4 * lane + k] = acc[4 + k];
    }
    __syncthreads();

    float ssum = 0.0f;
#pragma unroll
    for (int ww = 0; ww < 8; ++ww) ssum += red[ww][tid];   // fixed order
    part[((size_t)s * P + pb) * CDIM + tid] = ssum;
}

// ---------------------------------------------------------------------------
// Kernel 3: reduce P partials per segment (fixed ascending order), then
// g[s][c] = (sum / count)^(1/p).  grid = 32 blocks, block = 256.
// ---------------------------------------------------------------------------
__global__ void k_reduce_g(const float* __restrict__ part, const int* __restrict__ start,
                           const float* __restrict__ p, float* __restrict__ g, int P) {
    const int s = blockIdx.x;
    const int t = threadIdx.x;
    float sum = 0.0f;
    for (int pb = 0; pb < P; ++pb)
        sum += part[((size_t)s * P + pb) * CDIM + t];
    const int cnt = start[s + 1] - start[s];
    float gv = 0.0f;
    if (cnt > 0) {
        float mean = sum / (float)cnt;
        float ip = 1.0f / p[0];
        gv = __builtin_amdgcn_exp2f(ip * __builtin_amdgcn_logf(mean));
    }
    g[s * CDIM + t] = gv;
}

// ---------------------------------------------------------------------------
// Kernel 4: out_raw = g @ W^T + b via V_WMMA_F32_16X16X4_F32.
// grid = 4 blocks x 256 threads = 32 waves; wave -> 16x16 tile (mt, nt),
// K-loop of 64 accumulating WMMAs (K=4 each).
// A (16x4 f32): lanes 0-15 hold M=lane, VGPRs = K {kb, kb+1}; lanes 16-31 K {kb+2, kb+3}.
// B (4x16 f32): lanes hold N, same K split; B[k][n] = W[n][k].
// D: VGPR v -> M = v (lanes<16) or v+8 (lanes>=16), N = lane&15.
// ---------------------------------------------------------------------------
__global__ void k_gemm_wmma(const float* __restrict__ g, const float* __restrict__ W,
                            const float* __restrict__ bias, float* __restrict__ out) {
    const int tid   = threadIdx.x;
    const int lane  = tid & 31;
    const int wave  = blockIdx.x * 8 + (tid >> 5);   // 0..31
    const int mt    = wave >> 4;                     // 0..1
    const int nt    = wave & 15;                     // 0..15

    const int m  = mt * 16 + (lane & 15);
    const int n  = nt * 16 + (lane & 15);
    const int kh = (lane >> 4) << 1;                 // 0 or 2

    v8f c = {0.f, 0.f, 0.f, 0.f, 0.f, 0.f, 0.f, 0.f};
    const float* grow = g + (size_t)m * CDIM;
    const float* wrow = W + (size_t)n * CDIM;

#pragma unroll 4
    for (int kb = 0; kb < CDIM; kb += 4) {
        v2f a = *(const v2f*)(grow + kb + kh);
        v2f bv = *(const v2f*)(wrow + kb + kh);
        c = __builtin_amdgcn_wmma_f32_16x16x4_f32(
                /*neg_a=*/false, a, /*neg_b=*/false, bv,
                /*c_mod=*/(short)0, c, /*reuse_a=*/false, /*reuse_b=*/false);
    }

    const int col   = nt * 16 + (lane & 15);
    const int rbase = mt * 16 + ((lane >> 4) << 3);
    const float bb  = bias[col];
#pragma unroll
    for (int v = 0; v < 8; ++v)
        out[(size_t)(rbase + v) * OUT_DIM + col] = c[v] + bb;
}

// ---------------------------------------------------------------------------
// Kernel 5: in-place row L2 normalize. grid = 32 blocks, block = 256.
// ---------------------------------------------------------------------------
__global__ void k_normalize(float* __restrict__ out) {
    const int row = blockIdx.x;
    const int t   = threadIdx.x;
    float v = out[(size_t)row * OUT_DIM + t];
    __shared__ float red[OUT_DIM];
    red[t] = v * v;
    __syncthreads();
    for (int s = OUT_DIM / 2; s > 0; s >>= 1) {
        if (t < s) red[t] += red[t + s];
        __syncthreads();
    }
    float inv = 1.0f / fmaxf(sqrtf(red[0]), 1e-12f);
    out[(size_t)row * OUT_DIM + t] = v * inv;
}

// ---------------------------------------------------------------------------
// Host launch
// ---------------------------------------------------------------------------
extern "C" void kernel_launch(void* const* d_in, const int* in_sizes, int n_in,
                              void* d_out, int out_size, void* d_ws, size_t ws_size,
                              hipStream_t stream) {
    (void)n_in; (void)out_size;
    const float* x   = (const float*)d_in[0];
    const float* p   = (const float*)d_in[1];
    const float* W   = (const float*)d_in[2];
    const float* b   = (const float*)d_in[3];
    const int*   idx = (const int*)d_in[4];
    float* out = (float*)d_out;

    const int N = in_sizes[0] / CDIM;

    // Workspace layout (floats): [0..63] segment starts (33 ints used),
    // then part[NSEG][P][CDIM], then g[NSEG][CDIM].
    size_t avail = ws_size / sizeof(float);
    int P = 64;  // blocks per segment for the streaming reduction
    while (P > 1 && (64 + (size_t)NSEG * P * CDIM + NSEG * CDIM) > avail) P >>= 1;

    int*   start = (int*)d_ws;
    float* part  = (float*)d_ws + 64;
    float* g     = part + (size_t)NSEG * P * CDIM;

    k_boundaries<<<1, 64, 0, stream>>>(idx, N, start);
    k_partials<<<dim3(P, NSEG), 256, 0, stream>>>(x, p, start, part, P);
    k_reduce_g<<<NSEG, 256, 0, stream>>>(part, start, p, g, P);
    k_gemm_wmma<<<4, 256, 0, stream>>>(g, W, b, out);
    k_normalize<<<NSEG, 256, 0, stream>>>(out);
}